// MyLoss2_80736795230576
// MI455X (gfx1250) — compile-verified
//
#include <hip/hip_runtime.h>

#define NUM_CLASSES 10
#define BINS 16          // classes padded to 16 (rows of WMMA M dim); id==10 is the dummy bin
#define BLOCK 256        // 8 wave32 per block
#define NWAVES 8

typedef __attribute__((ext_vector_type(2))) float v2f;
typedef __attribute__((ext_vector_type(4))) float v4f;
typedef __attribute__((ext_vector_type(4))) int   v4i;
typedef __attribute__((ext_vector_type(8))) float v8f;

// ws layout: ws[0..9] per-class sqerr sums, ws[10..19] per-class counts (as float)
__global__ void init_ws_kernel(float* ws) {
    int t = threadIdx.x;
    if (t < 2 * NUM_CLASSES) ws[t] = 0.0f;
}

__global__ __launch_bounds__(BLOCK) void bin_kernel(const float* __restrict__ outp,
                                                    const float* __restrict__ tgtp,
                                                    const int*   __restrict__ mskp,
                                                    float* __restrict__ ws,
                                                    long long n4) {
    __shared__ float lsum[NWAVES][BINS];
    __shared__ float lcnt[NWAVES][BINS];

    const int tid  = threadIdx.x;
    const int wave = tid >> 5;
    const int lane = tid & 31;

    if (tid < NWAVES * BINS) {
        (&lsum[0][0])[tid] = 0.0f;
        (&lcnt[0][0])[tid] = 0.0f;
    }
    __syncthreads();

    const v4f* __restrict__ o4 = (const v4f*)outp;
    const v4f* __restrict__ t4 = (const v4f*)tgtp;
    const v4i* __restrict__ m4 = (const v4i*)mskp;

    const long long stride = (long long)gridDim.x * BLOCK;
    for (long long i = (long long)blockIdx.x * BLOCK + tid; i < n4; i += stride) {
        // streaming data: non-temporal B128 loads (TH_NT) — don't thrash L2
        v4f o = __builtin_nontemporal_load(o4 + i);
        v4f t = __builtin_nontemporal_load(t4 + i);
        v4i m = __builtin_nontemporal_load(m4 + i);

        float d0 = o.x - t.x, d1 = o.y - t.y, d2 = o.z - t.z, d3 = o.w - t.w;
        float e0 = d0 * d0, e1 = d1 * d1, e2 = d2 * d2, e3 = d3 * d3;

        int id0 = (m.x == 1) ? (int)t.x : NUM_CLASSES;
        int id1 = (m.y == 1) ? (int)t.y : NUM_CLASSES;
        int id2 = (m.z == 1) ? (int)t.z : NUM_CLASSES;
        int id3 = (m.w == 1) ? (int)t.w : NUM_CLASSES;

        // per-wave LDS bins: ds_add_f32 atomics
        unsafeAtomicAdd(&lsum[wave][id0], e0); unsafeAtomicAdd(&lcnt[wave][id0], 1.0f);
        unsafeAtomicAdd(&lsum[wave][id1], e1); unsafeAtomicAdd(&lcnt[wave][id1], 1.0f);
        unsafeAtomicAdd(&lsum[wave][id2], e2); unsafeAtomicAdd(&lcnt[wave][id2], 1.0f);
        unsafeAtomicAdd(&lsum[wave][id3], e3); unsafeAtomicAdd(&lcnt[wave][id3], 1.0f);
    }
    __syncthreads();

    // Cross-wave block reduction via full-precision matrix op:
    // D[m][n] += A[m][k] * B[k][n]; A = bins (16 classes x 4 waves, f32),
    // B one-hot column selector (n==0 -> sums, n==1 -> counts). 4 WMMAs / block.
    if (tid < 32) {
        const int mrow  = lane & 15;           // A row (class) / B column index n
        const int kbase = (lane < 16) ? 0 : 2; // A K-slot pair held by this lane

        const float selN0 = (mrow == 0) ? 1.0f : 0.0f;
        const float selN1 = (mrow == 1) ? 1.0f : 0.0f;
        v2f b_sum = { selN0, selN0 };   // B[k][n] = (n==0)
        v2f b_cnt = { selN1, selN1 };   // B[k][n] = (n==1)

        v2f a_s03 = { lsum[kbase + 0][mrow], lsum[kbase + 1][mrow] };
        v2f a_s47 = { lsum[4 + kbase + 0][mrow], lsum[4 + kbase + 1][mrow] };
        v2f a_c03 = { lcnt[kbase + 0][mrow], lcnt[kbase + 1][mrow] };
        v2f a_c47 = { lcnt[4 + kbase + 0][mrow], lcnt[4 + kbase + 1][mrow] };

        v8f c = {};
        c = __builtin_amdgcn_wmma_f32_16x16x4_f32(false, a_s03, false, b_sum, (short)0, c, false, false);
        c = __builtin_amdgcn_wmma_f32_16x16x4_f32(false, a_s47, false, b_sum, (short)0, c, false, false);
        c = __builtin_amdgcn_wmma_f32_16x16x4_f32(false, a_c03, false, b_cnt, (short)0, c, false, false);
        c = __builtin_amdgcn_wmma_f32_16x16x4_f32(false, a_c47, false, b_cnt, (short)0, c, false, false);

        // D layout (16x16 f32 C/D): VGPR r, lanes 0-15 -> M=r,N=lane; lanes 16-31 -> M=r+8,N=lane-16
        if (lane == 0) {            // sums, classes 0..7
#pragma unroll
            for (int r = 0; r < 8; ++r) unsafeAtomicAdd(&ws[r], c[r]);
        } else if (lane == 16) {    // sums, classes 8..9
            unsafeAtomicAdd(&ws[8], c[0]);
            unsafeAtomicAdd(&ws[9], c[1]);
        } else if (lane == 1) {     // counts, classes 0..7
#pragma unroll
            for (int r = 0; r < 8; ++r) unsafeAtomicAdd(&ws[10 + r], c[r]);
        } else if (lane == 17) {    // counts, classes 8..9
            unsafeAtomicAdd(&ws[18], c[0]);
            unsafeAtomicAdd(&ws[19], c[1]);
        }
    }
}

// out layout: [loss, loss_each(10), class_n(10)] = 21 floats
__global__ void finalize_kernel(const float* __restrict__ ws, float* __restrict__ out) {
    if (threadIdx.x == 0) {
        float loss = 0.0f;
#pragma unroll
        for (int cN = 0; cN < NUM_CLASSES; ++cN) {
            float s = ws[cN];
            float n = ws[10 + cN];
            float le = (n > 0.0f) ? (s / fmaxf(n, 1.0f)) : 0.0f;
            out[1 + cN]  = le;
            out[11 + cN] = n;
            loss += 0.1f * le;
        }
        out[0] = loss;
    }
}

extern "C" void kernel_launch(void* const* d_in, const int* in_sizes, int n_in,
                              void* d_out, int out_size, void* d_ws, size_t ws_size,
                              hipStream_t stream) {
    const float* outp = (const float*)d_in[0];
    const float* tgtp = (const float*)d_in[1];
    const int*   mskp = (const int*)d_in[2];
    float* ws  = (float*)d_ws;
    float* out = (float*)d_out;

    long long n  = (long long)in_sizes[0];   // 64*512*512, multiple of 4
    long long n4 = n >> 2;

    init_ws_kernel<<<1, 32, 0, stream>>>(ws);

    long long wantBlocks = (n4 + BLOCK - 1) / BLOCK;
    int blocks = (int)(wantBlocks < 4096 ? wantBlocks : 4096);
    bin_kernel<<<blocks, BLOCK, 0, stream>>>(outp, tgtp, mskp, ws, n4);

    finalize_kernel<<<1, 32, 0, stream>>>(ws, out);
}